// DynamicsAttention_17205638988011
// MI455X (gfx1250) — compile-verified
//
#include <hip/hip_runtime.h>
#include <math.h>

// ---------------------------------------------------------------------------
// DynamicsAttention for MI455X (gfx1250), fp32 end-to-end.
// - All GEMMs (precompute v/k, scan logits/r/GRU/q-update, final MLP) run on
//   V_WMMA_F32_16X16X4_F32 (exact f32 numerics, matches the f32 reference).
// - The x-tile for the big precompute GEMM is DMA'd into LDS by the Tensor
//   Data Mover (tensor_load_to_lds + s_wait_tensorcnt).
// ---------------------------------------------------------------------------

#define BSZ   32
#define TT    32
#define CDIM  128
#define HALFC 64
#define WH    256      // 16*16 pixels
#define NOBJ  16
#define ODIM  64

typedef __attribute__((ext_vector_type(2))) float v2f;
typedef __attribute__((ext_vector_type(8))) float v8f;
typedef __attribute__((ext_vector_type(4))) unsigned int u32x4;
typedef __attribute__((ext_vector_type(8))) int i32x8;
typedef __attribute__((ext_vector_type(4))) int i32x4;

__device__ __forceinline__ v8f wmma4(v2f a, v2f b, v8f c) {
  return __builtin_amdgcn_wmma_f32_16x16x4_f32(false, a, false, b, (short)0, c,
                                               false, false);
}

// -------------------------------------------------------------------- pe ----
__global__ void pe_kernel(const float* __restrict__ pe_w,
                          const float* __restrict__ pe_b,
                          float* __restrict__ pe) {
  int idx = blockIdx.x * blockDim.x + threadIdx.x;           // 128*256
  if (idx >= CDIM * WH) return;
  int c = idx >> 8, p = idx & 255;
  int i = p >> 4, j = p & 15;
  float fi = (float)i * (1.0f / 15.0f);
  float fj = (float)j * (1.0f / 15.0f);
  pe[idx] = pe_w[c * 4 + 0] * fi + pe_w[c * 4 + 1] * fj +
            pe_w[c * 4 + 2] * (1.0f - fi) + pe_w[c * 4 + 3] * (1.0f - fj) +
            pe_b[c];
}

// pe_v[o][p] = v_b[o] + sum_c v_w[o][c]*pe[c][p]   (constant over b,t)
__global__ void pev_kernel(const float* __restrict__ pe,
                           const float* __restrict__ v_w,
                           const float* __restrict__ v_b,
                           float* __restrict__ pe_v) {
  int idx = blockIdx.x * blockDim.x + threadIdx.x;
  if (idx >= CDIM * WH) return;
  int o = idx >> 8, p = idx & 255;
  float s = v_b[o];
  for (int c = 0; c < CDIM; ++c) s += v_w[o * CDIM + c] * pe[c * WH + p];
  pe_v[idx] = s;
}

// ----------------------------------------------------------- v / k / beta ---
// One block per (bt, 64-pixel tile). x-tile DMA'd into LDS by the TDM,
// 8 waves x 6 WMMA tiles.
__global__ void vkb_kernel(const float* __restrict__ x,
                           const float* __restrict__ v_w,
                           const float* __restrict__ k_w,
                           const float* __restrict__ k_b,
                           const float* __restrict__ beta_w,
                           const float* __restrict__ beta_b,
                           const float* __restrict__ pe_v,
                           float* __restrict__ v_out,
                           float* __restrict__ k_out,
                           float* __restrict__ beta_out) {
  __shared__ __align__(16) float xs[CDIM * 64];
  const int bt  = blockIdx.x;          // b*T + t (same order as inputs)
  const int p0  = blockIdx.y * 64;     // pixel tile base
  const int tid = threadIdx.x;
  const float* xin = x + (size_t)bt * CDIM * WH;

  // --- Tensor Data Mover: DMA 128 rows x 64 floats (row stride 256) into LDS
  // D# per CDNA5 ISA ch.8: group0 {count, lds_addr, global_addr[56:0], type=2}
  //                        group1 {data_size=4B, dims/tile/stride}
  if (tid < 32) {  // one wave issues the descriptor (EXEC-independent DMA)
    unsigned long long ga = (unsigned long long)(uintptr_t)(xin + p0);
    unsigned lds_off = (unsigned)(uintptr_t)(&xs[0]);  // low 32 = LDS offset
    u32x4 g0;
    g0[0] = 1u;                                   // count=1 valid descriptor
    g0[1] = lds_off;                              // lds_addr (bytes)
    g0[2] = (unsigned)ga;                         // global_addr[31:0]
    g0[3] = (unsigned)(ga >> 32) | 0x80000000u;   // addr[56:32] | type=2
    i32x8 g1;
    g1[0] = 0x00020000;     // workgroup_mask=0, data_size=2 (4 bytes)
    g1[1] = 0x01000000;     // tensor_dim0[15:0]=256 in bits[31:16]
    g1[2] = 0x00800000;     // tensor_dim1[15:0]=128 in bits[31:16]
    g1[3] = 0x00400000;     // tile_dim0=64 in bits[31:16]
    g1[4] = 128;            // tile_dim1=128
    g1[5] = 256;            // tensor_dim0_stride=256
    g1[6] = 0;
    g1[7] = 0;
    i32x4 gz4 = {};
    i32x8 gz8 = {};
    __builtin_amdgcn_tensor_load_to_lds(g0, g1, gz4, gz4, gz8, 0);
    __builtin_amdgcn_s_wait_tensorcnt(0);
  }
  __syncthreads();

  const int wave = tid >> 5, lane = tid & 31;
  const int half = lane >> 4, l16 = lane & 15;

  // tiles 0..31 : v (8 M-tiles x 4 N-tiles), 32..47 : k (4 x 4)
  for (int tile = wave; tile < 48; tile += 8) {
    const bool isV = tile < 32;
    const int  tt  = isV ? tile : tile - 32;
    const int  m0  = (tt >> 2) * 16;
    const int  n0  = (tt & 3) * 16;
    const float* W = isV ? v_w : k_w;

    v8f acc = {};
    for (int k0 = 0; k0 < CDIM; k0 += 4) {
      const int kb = k0 + half * 2;                   // K pair per lane-half
      v2f a, b;
      a.x = W[(m0 + l16) * CDIM + kb];                // A: 16x4 of weights
      a.y = W[(m0 + l16) * CDIM + kb + 1];
      b.x = xs[kb * 64 + n0 + l16];                   // B: 4x16 of x
      b.y = xs[(kb + 1) * 64 + n0 + l16];
      acc = wmma4(a, b, acc);
    }
    const int col = p0 + n0 + l16;
    if (isV) {
      float* dst = v_out + (size_t)bt * CDIM * WH;
#pragma unroll
      for (int r = 0; r < 8; ++r) {
        int row = m0 + half * 8 + r;
        dst[row * WH + col] = acc[r] + pe_v[row * WH + col];
      }
    } else {
      float* dst = k_out + (size_t)bt * HALFC * WH;
#pragma unroll
      for (int r = 0; r < 8; ++r) {
        int row = m0 + half * 8 + r;
        dst[row * WH + col] = acc[r] + k_b[row];
      }
    }
  }

  // beta = softplus(b) + 1 (stable), 64 pixels per block
  if (tid < 64) {
    float s = beta_b[0];
    for (int c = 0; c < CDIM; ++c) s += beta_w[c] * xs[c * 64 + tid];
    float bv = fmaxf(s, 0.0f) + log1pf(expf(-fabsf(s))) + 1.0f;
    beta_out[(size_t)bt * WH + p0 + tid] = bv;
  }
}

// --------------------------------------------------------- conv (stride 2) --
__global__ void conv_s2_kernel(const float* __restrict__ in,
                               const float* __restrict__ w,
                               const float* __restrict__ bias,
                               float* __restrict__ out,
                               int Cin, int Cout, int Hin, int Hout,
                               int in_batch_stride, int n_total) {
  int idx = blockIdx.x * blockDim.x + threadIdx.x;
  if (idx >= n_total) return;
  int ox = idx % Hout;
  int oy = (idx / Hout) % Hout;
  int oc = (idx / (Hout * Hout)) % Cout;
  int b  = idx / (Hout * Hout * Cout);
  float acc = bias[oc];
  for (int ic = 0; ic < Cin; ++ic) {
    const float* ip = in + (size_t)b * in_batch_stride + ic * Hin * Hin;
    const float* wp = w + ((size_t)oc * Cin + ic) * 16;
    for (int ky = 0; ky < 4; ++ky) {
      int iy = oy * 2 - 1 + ky;
      if (iy < 0 || iy >= Hin) continue;
      for (int kx = 0; kx < 4; ++kx) {
        int ix = ox * 2 - 1 + kx;
        if (ix < 0 || ix >= Hin) continue;
        acc += ip[iy * Hin + ix] * wp[ky * 4 + kx];
      }
    }
  }
  out[idx] = acc;
}

// ------------------------------------------------------------------- scan ---
// One persistent block per batch; q/h live in LDS across the 32 steps.
// All four per-step GEMMs are f32 WMMA (M=16 == N_OBJ => one tile row each).
__global__ void scan_kernel(const float* __restrict__ vv,
                            const float* __restrict__ kk,
                            const float* __restrict__ betas,
                            const float* __restrict__ q0,   // conv4 out [b][1024]
                            const float* __restrict__ q_w,
                            const float* __restrict__ q_b,
                            const float* __restrict__ wih,
                            const float* __restrict__ whh,
                            const float* __restrict__ bih,
                            const float* __restrict__ bhh,
                            float* __restrict__ outs,        // [bt][16][128]
                            float* __restrict__ attns_out) { // (B,256,T,16)
  __shared__ float q[NOBJ * HALFC];
  __shared__ float hA[NOBJ * CDIM];
  __shared__ float hB[NOBJ * CDIM];
  __shared__ float attn[NOBJ * WH];
  __shared__ float rbuf[NOBJ * CDIM];
  __shared__ float red[256];
  __shared__ float rowred[NOBJ];
  const int b = blockIdx.x;
  const int tid = threadIdx.x;
  const int wave = tid >> 5, lane = tid & 31;
  const int half = lane >> 4, l16 = lane & 15;

  for (int i = tid; i < NOBJ * HALFC; i += 256) {
    int n = i >> 6, o = i & 63;
    q[i] = q0[b * 1024 + o * NOBJ + n];    // reshape(64,16).T
  }
  for (int i = tid; i < NOBJ * CDIM; i += 256) hA[i] = 0.0f;
  __syncthreads();

  for (int t = 0; t < TT; ++t) {
    const size_t bt = (size_t)b * TT + t;
    const float* kt    = kk    + bt * HALFC * WH;
    const float* vt    = vv    + bt * CDIM * WH;
    const float* betat = betas + bt * WH;
    float* hOld = (t & 1) ? hB : hA;
    float* hNew = (t & 1) ? hA : hB;

    // ---- logits = q (16x64) @ k (64x256): 16 N-tiles, 2 per wave (WMMA)
#pragma unroll
    for (int rep = 0; rep < 2; ++rep) {
      const int mpix0 = (wave * 2 + rep) * 16;
      v8f acc = {};
      for (int k0 = 0; k0 < HALFC; k0 += 4) {
        const int kb = k0 + half * 2;
        v2f a, bf;
        a.x  = q[l16 * HALFC + kb];
        a.y  = q[l16 * HALFC + kb + 1];
        bf.x = kt[kb * WH + mpix0 + l16];
        bf.y = kt[(kb + 1) * WH + mpix0 + l16];
        acc = wmma4(a, bf, acc);
      }
#pragma unroll
      for (int r = 0; r < 8; ++r) {
        int n = r + half * 8;
        attn[n * WH + mpix0 + l16] = acc[r];
      }
    }
    __syncthreads();

    // ---- row softmax * beta + eps (16 threads per row, scalar)
    {
      int n = tid >> 4, seg = tid & 15;
      float lmax = -3.4e38f;
      for (int j = 0; j < 16; ++j)
        lmax = fmaxf(lmax, attn[n * WH + seg * 16 + j]);
      red[tid] = lmax;
      __syncthreads();
      if (tid < NOBJ) {
        float m = red[tid * 16];
        for (int j = 1; j < 16; ++j) m = fmaxf(m, red[tid * 16 + j]);
        rowred[tid] = m;
      }
      __syncthreads();
      float rmax = rowred[n];
      float lsum = 0.0f;
      for (int j = 0; j < 16; ++j) {
        int m = seg * 16 + j;
        float e = __expf(attn[n * WH + m] - rmax);
        attn[n * WH + m] = e;
        lsum += e;
      }
      red[tid] = lsum;
      __syncthreads();
      if (tid < NOBJ) {
        float s = 0.0f;
        for (int j = 0; j < 16; ++j) s += red[tid * 16 + j];
        rowred[tid] = s;
      }
      __syncthreads();
      float rinv = 1.0f / rowred[n];
      for (int j = 0; j < 16; ++j) {
        int m = seg * 16 + j;
        float a = attn[n * WH + m] * rinv * betat[m] + 1e-8f;
        attn[n * WH + m] = a;
        attns_out[(((size_t)b * WH + m) * TT + t) * NOBJ + n] = a;
      }
    }
    __syncthreads();

    // ---- r = attn (16x256) @ v^T (256x128): 8 c-tiles, 1 per wave (WMMA)
    {
      const int c0 = wave * 16;
      v8f acc = {};
      for (int k0 = 0; k0 < WH; k0 += 4) {
        const int kb = k0 + half * 2;
        v2f a, bf;
        a.x  = attn[l16 * WH + kb];
        a.y  = attn[l16 * WH + kb + 1];
        bf.x = vt[(c0 + l16) * WH + kb];     // B[k=m][n=c] = v[c][m]
        bf.y = vt[(c0 + l16) * WH + kb + 1];
        acc = wmma4(a, bf, acc);
      }
#pragma unroll
      for (int r = 0; r < 8; ++r) {
        int n = r + half * 8;
        rbuf[n * CDIM + c0 + l16] = acc[r];
      }
    }
    __syncthreads();

    // ---- GRU: per wave one 16-wide c-slice; 6 WMMA accumulators fuse the
    //      r/z/n gates for both the input (r) and hidden (h) paths.
    {
      const int c0 = wave * 16;
      v8f air = {}, aiz = {}, ain = {}, ahr = {}, ahz = {}, ahn = {};
      for (int k0 = 0; k0 < CDIM; k0 += 4) {
        const int kb = k0 + half * 2;
        v2f ar, ah, br, bz, bn, ur, uz, un;
        ar.x = rbuf[l16 * CDIM + kb];  ar.y = rbuf[l16 * CDIM + kb + 1];
        ah.x = hOld[l16 * CDIM + kb];  ah.y = hOld[l16 * CDIM + kb + 1];
        br.x = wih[(size_t)(c0 + l16) * CDIM + kb];
        br.y = wih[(size_t)(c0 + l16) * CDIM + kb + 1];
        bz.x = wih[(size_t)(128 + c0 + l16) * CDIM + kb];
        bz.y = wih[(size_t)(128 + c0 + l16) * CDIM + kb + 1];
        bn.x = wih[(size_t)(256 + c0 + l16) * CDIM + kb];
        bn.y = wih[(size_t)(256 + c0 + l16) * CDIM + kb + 1];
        ur.x = whh[(size_t)(c0 + l16) * CDIM + kb];
        ur.y = whh[(size_t)(c0 + l16) * CDIM + kb + 1];
        uz.x = whh[(size_t)(128 + c0 + l16) * CDIM + kb];
        uz.y = whh[(size_t)(128 + c0 + l16) * CDIM + kb + 1];
        un.x = whh[(size_t)(256 + c0 + l16) * CDIM + kb];
        un.y = whh[(size_t)(256 + c0 + l16) * CDIM + kb + 1];
        air = wmma4(ar, br, air);
        aiz = wmma4(ar, bz, aiz);
        ain = wmma4(ar, bn, ain);
        ahr = wmma4(ah, ur, ahr);
        ahz = wmma4(ah, uz, ahz);
        ahn = wmma4(ah, un, ahn);
      }
      const int c = c0 + l16;
#pragma unroll
      for (int r = 0; r < 8; ++r) {
        int n = r + half * 8;
        float ir  = air[r] + bih[c];
        float hr  = ahr[r] + bhh[c];
        float iz  = aiz[r] + bih[128 + c];
        float hz  = ahz[r] + bhh[128 + c];
        float inn = ain[r] + bih[256 + c];
        float hnn = ahn[r] + bhh[256 + c];
        float rg = 1.0f / (1.0f + __expf(-(ir + hr)));
        float z  = 1.0f / (1.0f + __expf(-(iz + hz)));
        float ng = tanhf(inn + rg * hnn);
        float hv = (1.0f - z) * ng + z * hOld[n * CDIM + c];
        hNew[n * CDIM + c] = hv;
        outs[bt * (NOBJ * CDIM) + n * CDIM + c] = hv;
      }
    }
    __syncthreads();

    // ---- q = hNew (16x128) @ q_w^T (128x64) + q_b : 4 N-tiles, waves 0-3
    if (wave < 4) {           // wave-uniform branch: EXEC all-ones inside
      const int o0 = wave * 16;
      v8f acc = {};
      for (int k0 = 0; k0 < CDIM; k0 += 4) {
        const int kb = k0 + half * 2;
        v2f a, bf;
        a.x  = hNew[l16 * CDIM + kb];
        a.y  = hNew[l16 * CDIM + kb + 1];
        bf.x = q_w[(size_t)(o0 + l16) * CDIM + kb];
        bf.y = q_w[(size_t)(o0 + l16) * CDIM + kb + 1];
        acc = wmma4(a, bf, acc);
      }
      const int o = o0 + l16;
#pragma unroll
      for (int r = 0; r < 8; ++r) {
        int n = r + half * 8;
        q[n * HALFC + o] = acc[r] + q_b[o];
      }
    }
    __syncthreads();
  }
}

// ----------------------------------------------------------- final MLP ------
// per (b,t): frames[b][n][t][o] = outs[bt][n] . mlp_w[o] + mlp_b[o]
// M=16, N=64, K=128; 4 waves = 4 N-tiles of f32 WMMA.
__global__ void mlp_kernel(const float* __restrict__ outs,
                           const float* __restrict__ mlp_w,
                           const float* __restrict__ mlp_b,
                           float* __restrict__ frames) {
  const int bt = blockIdx.x;
  const int b = bt >> 5, t = bt & 31;
  const int wave = threadIdx.x >> 5, lane = threadIdx.x & 31;
  const int half = lane >> 4, l16 = lane & 15;
  const int n0 = wave * 16;
  const float* S = outs + (size_t)bt * NOBJ * CDIM;

  v8f acc = {};
  for (int k0 = 0; k0 < CDIM; k0 += 4) {
    int kb = k0 + half * 2;
    v2f a, bf;
    a.x  = S[l16 * CDIM + kb];                // A: 16x4 of h rows
    a.y  = S[l16 * CDIM + kb + 1];
    bf.x = mlp_w[(n0 + l16) * CDIM + kb];     // B[k][o] = mlp_w[o][k]
    bf.y = mlp_w[(n0 + l16) * CDIM + kb + 1];
    acc = wmma4(a, bf, acc);
  }
  const int o = n0 + l16;
#pragma unroll
  for (int r = 0; r < 8; ++r) {
    int n = r + half * 8;
    frames[(((size_t)b * NOBJ + n) * TT + t) * ODIM + o] = acc[r] + mlp_b[o];
  }
}

// ------------------------------------------------------------------ launch --
extern "C" void kernel_launch(void* const* d_in, const int* in_sizes, int n_in,
                              void* d_out, int out_size, void* d_ws,
                              size_t ws_size, hipStream_t stream) {
  (void)in_sizes; (void)n_in; (void)out_size; (void)ws_size;
  const float* inputs  = (const float*)d_in[0];
  const float* ini_w1  = (const float*)d_in[1];
  const float* ini_b1  = (const float*)d_in[2];
  const float* ini_w2  = (const float*)d_in[3];
  const float* ini_b2  = (const float*)d_in[4];
  const float* ini_w3  = (const float*)d_in[5];
  const float* ini_b3  = (const float*)d_in[6];
  const float* ini_w4  = (const float*)d_in[7];
  const float* ini_b4  = (const float*)d_in[8];
  const float* q_w     = (const float*)d_in[9];
  const float* q_b     = (const float*)d_in[10];
  const float* k_w     = (const float*)d_in[11];
  const float* k_b     = (const float*)d_in[12];
  const float* v_w     = (const float*)d_in[13];
  const float* v_b     = (const float*)d_in[14];
  const float* beta_w  = (const float*)d_in[15];
  const float* beta_b  = (const float*)d_in[16];
  const float* pe_w    = (const float*)d_in[17];
  const float* pe_b    = (const float*)d_in[18];
  const float* mlp_w   = (const float*)d_in[19];
  const float* mlp_b   = (const float*)d_in[20];
  const float* gru_wih = (const float*)d_in[21];
  const float* gru_whh = (const float*)d_in[22];
  const float* gru_bih = (const float*)d_in[23];
  const float* gru_bhh = (const float*)d_in[24];

  float* frames    = (float*)d_out;                       // (32,16,32,64)
  float* attns_out = frames + (size_t)BSZ * NOBJ * TT * ODIM;

  float* ws = (float*)d_ws;
  size_t off = 0;
  float* pe    = ws + off; off += (size_t)CDIM * WH;
  float* pe_v  = ws + off; off += (size_t)CDIM * WH;
  float* vbuf  = ws + off; off += (size_t)BSZ * TT * CDIM * WH;
  float* kbuf  = ws + off; off += (size_t)BSZ * TT * HALFC * WH;
  float* bbuf  = ws + off; off += (size_t)BSZ * TT * WH;
  float* cb1   = ws + off; off += (size_t)BSZ * 64 * 8 * 8;
  float* cb2   = ws + off; off += (size_t)BSZ * 64 * 4 * 4;
  float* cb3   = ws + off; off += (size_t)BSZ * 128 * 2 * 2;
  float* cb4   = ws + off; off += (size_t)BSZ * 1024;
  float* obuf  = ws + off; off += (size_t)BSZ * TT * NOBJ * CDIM;

  pe_kernel <<<128, 256, 0, stream>>>(pe_w, pe_b, pe);
  pev_kernel<<<128, 256, 0, stream>>>(pe, v_w, v_b, pe_v);

  dim3 vkb_grid(BSZ * TT, 4);
  vkb_kernel<<<vkb_grid, 256, 0, stream>>>(inputs, v_w, k_w, k_b, beta_w,
                                           beta_b, pe_v, vbuf, kbuf, bbuf);

  // conv encoder on inputs[:,0]
  {
    int n1 = BSZ * 64 * 8 * 8;
    conv_s2_kernel<<<(n1 + 255) / 256, 256, 0, stream>>>(
        inputs, ini_w1, ini_b1, cb1, 128, 64, 16, 8, TT * CDIM * WH, n1);
    int n2 = BSZ * 64 * 4 * 4;
    conv_s2_kernel<<<(n2 + 255) / 256, 256, 0, stream>>>(
        cb1, ini_w2, ini_b2, cb2, 64, 64, 8, 4, 64 * 64, n2);
    int n3 = BSZ * 128 * 2 * 2;
    conv_s2_kernel<<<(n3 + 255) / 256, 256, 0, stream>>>(
        cb2, ini_w3, ini_b3, cb3, 64, 128, 4, 2, 64 * 16, n3);
    int n4 = BSZ * 1024;
    conv_s2_kernel<<<(n4 + 255) / 256, 256, 0, stream>>>(
        cb3, ini_w4, ini_b4, cb4, 128, 1024, 2, 1, 128 * 4, n4);
  }

  scan_kernel<<<BSZ, 256, 0, stream>>>(vbuf, kbuf, bbuf, cb4, q_w, q_b,
                                       gru_wih, gru_whh, gru_bih, gru_bhh,
                                       obuf, attns_out);

  mlp_kernel<<<BSZ * TT, 128, 0, stream>>>(obuf, mlp_w, mlp_b, frames);
}